// GCN_76424648065474
// MI455X (gfx1250) — compile-verified
//
#include <hip/hip_runtime.h>
#include <math.h>

#define N_NODES 50000
#define N_EDGES 600000
#define NFEAT 128
#define NHID 128
#define NCLASS 41

typedef float v2f __attribute__((ext_vector_type(2)));
typedef float v8f __attribute__((ext_vector_type(8)));

// ---------------------------------------------------------------------------
// C[50000x128] = op(A[50000x128]) @ B[128x128], op = identity or relu(x + bias[k])
// One wave computes one 16x16 tile via V_WMMA_F32_16X16X4_F32 (fp32-exact).
// Block = 256 threads = 8 waves = one full row-tile across all 8 col-tiles.
// 50000 = 3125*16 exactly -> no remainder, EXEC all-ones (WMMA requirement).
//
// Fragment layouts per CDNA5 ISA 7.12.2 (wave32):
//   A 16x4 f32  : lane m=L%16 is row; half=L/16 selects K-pair; v0=K, v1=K+1 (contiguous float2)
//   B 4x16  f32 : lane n=L%16 is col; K = kb + 2*half + vgpr
//   C/D 16x16   : lane n=L%16 is col; row M = v + 8*half
// ---------------------------------------------------------------------------
template<bool BIAS_RELU>
__global__ __launch_bounds__(256)
void gemm128_wmma(const float* __restrict__ A, const float* __restrict__ B,
                  const float* __restrict__ bias, float* __restrict__ C)
{
    const int lane = threadIdx.x & 31;
    const int wave = threadIdx.x >> 5;     // 0..7 -> column tile
    const int m0   = blockIdx.x * 16;      // row tile base
    const int n0   = wave * 16;
    const int m    = lane & 15;            // row of A frag / col of B,C frag
    const int half = lane >> 4;

    const float* __restrict__ arow = A + (size_t)(m0 + m) * NHID;

    v8f acc = {};
#pragma unroll
    for (int kb = 0; kb < NHID; kb += 4) {
        const int ka = kb + 2 * half;
        float a0 = arow[ka];
        float a1 = arow[ka + 1];
        if (BIAS_RELU) {
            a0 = fmaxf(a0 + bias[ka],     0.0f);
            a1 = fmaxf(a1 + bias[ka + 1], 0.0f);
        }
        v2f a; a.x = a0; a.y = a1;
        v2f b;
        b.x = B[(size_t)ka       * NHID + n0 + m];
        b.y = B[(size_t)(ka + 1) * NHID + n0 + m];
        acc = __builtin_amdgcn_wmma_f32_16x16x4_f32(
                  /*neg_a=*/false, a, /*neg_b=*/false, b,
                  /*c_mod=*/(short)0, acc, /*reuse_a=*/false, /*reuse_b=*/false);
    }

#pragma unroll
    for (int v = 0; v < 8; ++v)
        C[(size_t)(m0 + v + 8 * half) * NHID + n0 + m] = acc[v];
}

// ---------------------------------------------------------------------------
// acc[dst] += w * h[src]  -- one wave per edge, float4 per lane (32*16B = 512B row).
// h (25.6MB) is L2-resident (192MB L2); atomics are native global_atomic_add_f32.
// ---------------------------------------------------------------------------
__global__ __launch_bounds__(256)
void spmm_atomic(const int* __restrict__ src, const int* __restrict__ dst,
                 const float* __restrict__ w, const float* __restrict__ h,
                 float* __restrict__ acc)
{
    const int e = (int)((blockIdx.x * blockDim.x + threadIdx.x) >> 5);
    if (e >= N_EDGES) return;
    const int lane = threadIdx.x & 31;
    const int s  = src[e];
    const int d  = dst[e];
    const float wt = w[e];

    const float4 hv = *((const float4*)(h + (size_t)s * NHID) + lane);
    float* ap = acc + (size_t)d * NHID + lane * 4;
    unsafeAtomicAdd(ap + 0, wt * hv.x);
    unsafeAtomicAdd(ap + 1, wt * hv.y);
    unsafeAtomicAdd(ap + 2, wt * hv.z);
    unsafeAtomicAdd(ap + 3, wt * hv.w);
}

__global__ __launch_bounds__(256)
void zero_f4(float4* __restrict__ p, int n4)
{
    const int i = blockIdx.x * blockDim.x + threadIdx.x;
    if (i < n4) p[i] = make_float4(0.f, 0.f, 0.f, 0.f);
}

// ---------------------------------------------------------------------------
// logits = relu(acc + b2) @ Wout + bout ; out = log_softmax(logits)
// One 64-thread block per node; 41 active class lanes; LDS reductions.
// ---------------------------------------------------------------------------
__global__ __launch_bounds__(64)
void head_logsoftmax(const float* __restrict__ acc, const float* __restrict__ b2,
                     const float* __restrict__ Wout, const float* __restrict__ bout,
                     float* __restrict__ out)
{
    __shared__ float hsh[NHID];
    __shared__ float red[64];
    const int node = blockIdx.x;
    const int t = threadIdx.x;

    for (int i = t; i < NHID; i += 64)
        hsh[i] = fmaxf(acc[(size_t)node * NHID + i] + b2[i], 0.0f);
    __syncthreads();

    float logit = -INFINITY;
    if (t < NCLASS) {
        float a = bout[t];
#pragma unroll 4
        for (int k = 0; k < NHID; ++k)
            a = fmaf(hsh[k], Wout[(size_t)k * NCLASS + t], a);
        logit = a;
    }

    // max reduction
    red[t] = logit;
    __syncthreads();
    for (int s = 32; s > 0; s >>= 1) {
        if (t < s) red[t] = fmaxf(red[t], red[t + s]);
        __syncthreads();
    }
    const float mx = red[0];
    __syncthreads();

    // sum(exp) reduction
    red[t] = (t < NCLASS) ? __expf(logit - mx) : 0.0f;
    __syncthreads();
    for (int s = 32; s > 0; s >>= 1) {
        if (t < s) red[t] += red[t + s];
        __syncthreads();
    }
    const float lse = __logf(red[0]) + mx;

    if (t < NCLASS)
        out[(size_t)node * NCLASS + t] = logit - lse;
}

// ---------------------------------------------------------------------------
extern "C" void kernel_launch(void* const* d_in, const int* in_sizes, int n_in,
                              void* d_out, int out_size, void* d_ws, size_t ws_size,
                              hipStream_t stream)
{
    const float* x    = (const float*)d_in[0];
    const int*   a0i  = (const int*)  d_in[1];   // [2, E] : row0=src, row1=dst
    const float* a0w  = (const float*)d_in[2];
    const int*   a1i  = (const int*)  d_in[3];
    const float* a1w  = (const float*)d_in[4];
    const float* W1   = (const float*)d_in[5];
    const float* b1   = (const float*)d_in[6];
    const float* W2   = (const float*)d_in[7];
    const float* b2   = (const float*)d_in[8];
    const float* Wout = (const float*)d_in[9];
    const float* bout = (const float*)d_in[10];
    float* out = (float*)d_out;

    const size_t NH = (size_t)N_NODES * NHID;   // 6.4M floats = 25.6 MB
    float* bufA = (float*)d_ws;                 // t0 / t1
    float* bufB = bufA + NH;                    // acc0 / acc1

    const dim3 gGemm(N_NODES / 16), bGemm(256);
    const int  n4 = (int)(NH / 4);
    const dim3 gZero((n4 + 255) / 256), bZero(256);
    const dim3 gSp((unsigned)(((size_t)N_EDGES * 32 + 255) / 256)), bSp(256);

    // layer 1: t0 = x @ W1 ; acc0 = A0 @ t0
    gemm128_wmma<false><<<gGemm, bGemm, 0, stream>>>(x, W1, nullptr, bufA);
    zero_f4<<<gZero, bZero, 0, stream>>>((float4*)bufB, n4);
    spmm_atomic<<<gSp, bSp, 0, stream>>>(a0i, a0i + N_EDGES, a0w, bufA, bufB);

    // layer 2: t1 = relu(acc0 + b1) @ W2 ; acc1 = A1 @ t1   (bias+relu fused into GEMM A-load)
    gemm128_wmma<true><<<gGemm, bGemm, 0, stream>>>(bufB, W2, b1, bufA);
    zero_f4<<<gZero, bZero, 0, stream>>>((float4*)bufB, n4);
    spmm_atomic<<<gSp, bSp, 0, stream>>>(a1i, a1i + N_EDGES, a1w, bufA, bufB);

    // head: log_softmax(relu(acc1 + b2) @ Wout + bout)
    head_logsoftmax<<<dim3(N_NODES), dim3(64), 0, stream>>>(bufB, b2, Wout, bout, out);
}